// Comp_Att_3358664425502
// MI455X (gfx1250) — compile-verified
//
#include <hip/hip_runtime.h>
#include <hip/hip_bf16.h>
#include <math.h>

typedef __attribute__((ext_vector_type(2))) float v2f;
typedef __attribute__((ext_vector_type(8))) float v8f;

#define NB 512          // batch
#define CH 1024         // channels
#define VJ 68           // joints
#define PP 7            // parts
#define OC 256          // C/4
#define MM (NB * PP)    // 3584 GEMM rows
#define KK CH           // GEMM K

// Part boundaries: [0,17) [17,22) [22,27) [27,36) [36,42) [42,48) [48,68)

template <int B, int E>
__device__ __forceinline__ float rsum(const float* __restrict__ row) {
    float s = 0.f;
#pragma unroll
    for (int v = B; v < E; ++v) s += row[v];
    return s;
}

// ---- k1: part-wise mean pooling  x(N,C,V) -> xp[(n*7+p)*1024 + c] ----------
__global__ __launch_bounds__(256) void k1_partpool(const float* __restrict__ x,
                                                   float* __restrict__ xp) {
    int tid = blockIdx.x * 256 + threadIdx.x;   // = n*CH + c
    int c = tid & (CH - 1);
    int n = tid >> 10;
    const float* row = x + (size_t)tid * VJ;
    size_t ob = (size_t)(n * PP) * CH + c;
    xp[ob + 0 * CH] = rsum< 0, 17>(row) * (1.f / 17.f);
    xp[ob + 1 * CH] = rsum<17, 22>(row) * (1.f / 5.f);
    xp[ob + 2 * CH] = rsum<22, 27>(row) * (1.f / 5.f);
    xp[ob + 3 * CH] = rsum<27, 36>(row) * (1.f / 9.f);
    xp[ob + 4 * CH] = rsum<36, 42>(row) * (1.f / 6.f);
    xp[ob + 5 * CH] = rsum<42, 48>(row) * (1.f / 6.f);
    xp[ob + 6 * CH] = rsum<48, 68>(row) * (1.f / 20.f);
}

// ---- k2: f32 WMMA GEMM  H[m,o] = sum_k xp[m,k] * W[o,k] + b1[o] ------------
// One wave per 16x16 tile. V_WMMA_F32_16X16X4_F32, K-loop of 256 steps.
// A 16x4 layout: lane = 16*hi + row holds A[m0+row, k+2*hi + {0,1}] in v[0..1]
// B  4x16 layout (mirror): lane holds B[k+2*hi + {0,1}, n0+row]; since
// B[k][n] = W[n][k], this is a contiguous float2 from W's row n0+row.
__global__ __launch_bounds__(128) void k2_gemm(const float* __restrict__ A,
                                               const float* __restrict__ W,
                                               const float* __restrict__ b1,
                                               float* __restrict__ H) {
    int tile = blockIdx.x * 4 + threadIdx.y;    // 3584 tiles
    int m0 = (tile >> 4) << 4;                  // 224 tiles along M
    int n0 = (tile & 15) << 4;                  // 16 tiles along O
    int lane = threadIdx.x;
    int row = lane & 15;
    int hi = lane >> 4;

    const float* arow = A + (size_t)(m0 + row) * KK + 2 * hi;
    const float* brow = W + (size_t)(n0 + row) * KK + 2 * hi;

    v8f acc = {};
    for (int k = 0; k < KK; k += 4) {
        v2f a = *(const v2f*)(arow + k);
        v2f b = *(const v2f*)(brow + k);
        acc = __builtin_amdgcn_wmma_f32_16x16x4_f32(
            false, a, false, b, (short)0, acc, false, false);
    }
    float bias = b1[n0 + row];
#pragma unroll
    for (int r = 0; r < 8; ++r) {
        // C/D layout: VGPR r, lanes 0-15 -> M=r, lanes 16-31 -> M=r+8
        H[(size_t)(m0 + r + 8 * hi) * OC + (n0 + row)] = acc[r] + bias;
    }
}

// ---- k3: per-row (m = n*7+p) mean + max over O=256, apply conv2 ------------
__global__ __launch_bounds__(256) void k3_reduce(const float* __restrict__ H,
                                                 const float* __restrict__ w2,
                                                 const float* __restrict__ b2,
                                                 float* __restrict__ attraw) {
    int m = blockIdx.x * 8 + (threadIdx.x >> 5);
    int lane = threadIdx.x & 31;
    const float* hr = H + (size_t)m * OC;
    float s = 0.f, mx = -INFINITY;
#pragma unroll
    for (int i = 0; i < 8; ++i) {
        float v = hr[lane + 32 * i];
        s += v;
        mx = fmaxf(mx, v);
    }
#pragma unroll
    for (int d = 16; d >= 1; d >>= 1) {
        s += __shfl_xor(s, d, 32);
        mx = fmaxf(mx, __shfl_xor(mx, d, 32));
    }
    if (lane == 0)
        attraw[m] = w2[0] * (s * (1.f / OC)) + w2[1] * mx + b2[0];
}

// ---- k3b: softmax over 7 parts per n, scatter to attv[n][v] ----------------
__device__ __forceinline__ void fillv(float* __restrict__ dst, int b, int e, float val) {
    for (int v = b; v < e; ++v) dst[v] = val;
}

__global__ __launch_bounds__(256) void k3b_softmax(const float* __restrict__ attraw,
                                                   float* __restrict__ attv) {
    int n = blockIdx.x * 256 + threadIdx.x;
    if (n >= NB) return;
    float a[PP];
    float mx = -INFINITY;
#pragma unroll
    for (int p = 0; p < PP; ++p) { a[p] = attraw[n * PP + p]; mx = fmaxf(mx, a[p]); }
    float s = 0.f;
#pragma unroll
    for (int p = 0; p < PP; ++p) { a[p] = __expf(a[p] - mx); s += a[p]; }
    float inv = 1.f / s;
#pragma unroll
    for (int p = 0; p < PP; ++p) a[p] *= inv;
    float* dst = attv + (size_t)n * VJ;
    fillv(dst,  0, 17, a[0]);
    fillv(dst, 17, 22, a[1]);
    fillv(dst, 22, 27, a[2]);
    fillv(dst, 27, 36, a[3]);
    fillv(dst, 36, 42, a[4]);
    fillv(dst, 42, 48, a[5]);
    fillv(dst, 48, 68, a[6]);
}

// ---- k4: per-channel BN stats of y = x*attv, fold into scale/shift ---------
__global__ __launch_bounds__(256) void k4_stats(const float* __restrict__ x,
                                                const float* __restrict__ attv,
                                                const float* __restrict__ gamma,
                                                const float* __restrict__ beta,
                                                float* __restrict__ scale,
                                                float* __restrict__ shift) {
    int c = blockIdx.x;
    int t = threadIdx.x;
    float s1 = 0.f, s2 = 0.f;
    for (int n = t; n < NB; n += 256) {
        const float4* xr = (const float4*)(x + ((size_t)n * CH + c) * VJ);
        const float4* ar = (const float4*)(attv + (size_t)n * VJ);
#pragma unroll
        for (int j = 0; j < 17; ++j) {
            float4 q = xr[j], w = ar[j];
            float y0 = q.x * w.x, y1 = q.y * w.y, y2 = q.z * w.z, y3 = q.w * w.w;
            s1 += (y0 + y1) + (y2 + y3);
            s2 += (y0 * y0 + y1 * y1) + (y2 * y2 + y3 * y3);
        }
    }
    __shared__ float sm1[256], sm2[256];
    sm1[t] = s1;
    sm2[t] = s2;
    __syncthreads();
    for (int d = 128; d >= 1; d >>= 1) {
        if (t < d) { sm1[t] += sm1[t + d]; sm2[t] += sm2[t + d]; }
        __syncthreads();
    }
    if (t == 0) {
        const float invNT = 1.f / (float)(NB * VJ);
        float mean = sm1[0] * invNT;
        float var = sm2[0] * invNT - mean * mean;
        float sc = gamma[c] * rsqrtf(var + 1e-5f);
        scale[c] = sc;
        shift[c] = beta[c] - mean * sc;
    }
}

// ---- k5: out = relu(x*(attv*scale + 1) + shift), float4 streaming ----------
__global__ __launch_bounds__(256) void k5_final(const float* __restrict__ x,
                                                const float* __restrict__ attv,
                                                const float* __restrict__ scale,
                                                const float* __restrict__ shift,
                                                float* __restrict__ out) {
    int idx = blockIdx.x * 256 + threadIdx.x;   // float4 index, 8912896 total
    int r = idx / 17;                           // row = n*CH + c
    int j = idx - r * 17;
    int c = r & (CH - 1);
    int n = r >> 10;
    float4 q = ((const float4*)x)[idx];
    float4 a = ((const float4*)(attv + (size_t)n * VJ))[j];
    float sc = scale[c], sh = shift[c];
    float4 o;
    o.x = fmaxf(fmaf(q.x, fmaf(a.x, sc, 1.f), sh), 0.f);
    o.y = fmaxf(fmaf(q.y, fmaf(a.y, sc, 1.f), sh), 0.f);
    o.z = fmaxf(fmaf(q.z, fmaf(a.z, sc, 1.f), sh), 0.f);
    o.w = fmaxf(fmaf(q.w, fmaf(a.w, sc, 1.f), sh), 0.f);
    ((float4*)out)[idx] = o;
}

extern "C" void kernel_launch(void* const* d_in, const int* in_sizes, int n_in,
                              void* d_out, int out_size, void* d_ws, size_t ws_size,
                              hipStream_t stream) {
    const float* x     = (const float*)d_in[0];   // (512,1024,1,68)
    const float* w1    = (const float*)d_in[1];   // (256,1024)
    const float* b1    = (const float*)d_in[2];   // (256,)
    const float* w2    = (const float*)d_in[3];   // (2,)
    const float* b2    = (const float*)d_in[4];   // ()
    const float* gamma = (const float*)d_in[5];   // (1024,)
    const float* beta  = (const float*)d_in[6];   // (1024,)
    float* out = (float*)d_out;

    float* ws     = (float*)d_ws;
    float* xp     = ws;                           // MM*CH   = 3,670,016 f
    float* H      = xp + (size_t)MM * CH;         // MM*OC   =   917,504 f
    float* attraw = H + (size_t)MM * OC;          // MM      =     3,584 f
    float* attv   = attraw + MM;                  // NB*VJ   =    34,816 f
    float* scale  = attv + (size_t)NB * VJ;       // CH
    float* shift  = scale + CH;                   // CH

    k1_partpool<<<(NB * CH) / 256, 256, 0, stream>>>(x, xp);
    k2_gemm<<<MM / 4, dim3(32, 4), 0, stream>>>(xp, w1, b1, H);
    k3_reduce<<<MM / 8, 256, 0, stream>>>(H, w2, b2, attraw);
    k3b_softmax<<<(NB + 255) / 256, 256, 0, stream>>>(attraw, attv);
    k4_stats<<<CH, 256, 0, stream>>>(x, attv, gamma, beta, scale, shift);
    k5_final<<<(NB * CH * VJ / 4) / 256, 256, 0, stream>>>(x, attv, scale, shift, out);
}